// Learnable_PPN_24893630448262
// MI455X (gfx1250) — compile-verified
//
#include <hip/hip_runtime.h>

#define N_NODES 100000
#define N_EDGES 1600000
#define DIMF    128
#define NCLS    100
#define LAYERS  3

typedef __attribute__((ext_vector_type(2))) float v2f;
typedef __attribute__((ext_vector_type(8))) float v8f;

// ---------------- utility: zero a float4 region ----------------
__global__ void zero_f4(float4* __restrict__ p, int n4) {
    int i = blockIdx.x * blockDim.x + threadIdx.x;
    if (i < n4) p[i] = make_float4(0.f, 0.f, 0.f, 0.f);
}

// ---------------- degree count: deg[dst[e]] += 1 ----------------
__global__ void deg_kernel(const int* __restrict__ dst, float* __restrict__ deg, int E) {
    int e = blockIdx.x * blockDim.x + threadIdx.x;
    if (e < E) atomicAdd(&deg[dst[e]], 1.0f);
}

// ---------------- deg -> deg^{-1/2} in place ----------------
__global__ void dis_kernel(float* __restrict__ deg, int n) {
    int i = blockIdx.x * blockDim.x + threadIdx.x;
    if (i < n) {
        float d = deg[i];
        deg[i] = (d > 0.0f) ? rsqrtf(d) : 0.0f;
    }
}

// ---------------- norm[e] = dis[src]*dis[dst] ----------------
__global__ void norm_kernel(const int* __restrict__ src, const int* __restrict__ dst,
                            const float* __restrict__ dis, float* __restrict__ norm, int E) {
    int e = blockIdx.x * blockDim.x + threadIdx.x;
    if (e < E) norm[e] = dis[src[e]] * dis[dst[e]];
}

// ---------------- seed: y0 = (mask ? onehot(label) : 0) @ protos via f32 WMMA ----------------
// Block = 256 threads = 8 wave32 waves. Block covers 128 nodes (16 per wave).
// Protos (100x128 f32 = 51.2KB) staged in LDS once per block.
// Exact: each WMMA product is 1.0*p or 0.0*q, so result == gather bit-for-bit.
__global__ __launch_bounds__(256) void seed_wmma(const unsigned char* __restrict__ mask,
                                                 const float* __restrict__ protos,
                                                 const int* __restrict__ labels,
                                                 float* __restrict__ y0,
                                                 float* __restrict__ out) {
    __shared__ float sh[NCLS * DIMF];
    int tid = threadIdx.x;
    for (int i = tid; i < NCLS * DIMF; i += 256) sh[i] = protos[i];
    __syncthreads();

    int wave = tid >> 5;
    int lane = tid & 31;
    int half = lane >> 4;   // K/M split across half-waves per ISA layout
    int n    = lane & 15;

    int nodeBase = blockIdx.x * 128 + wave * 16;

    // A-matrix row data for this lane (A is 16x4: lane&15 = M row, half selects K pair)
    int  node = nodeBase + n;
    bool inr  = node < N_NODES;
    int  nd   = inr ? node : (N_NODES - 1);
    int  lab  = labels[nd];
    bool tm   = inr && (mask[nd] != 0);

    v8f acc[8] = {};   // 8 tiles of 16x16 f32 covering D=128

    for (int k0 = 0; k0 < NCLS; k0 += 4) {
        int kb = k0 + half * 2;
        v2f a;
        a.x = (tm && lab == kb)     ? 1.0f : 0.0f;
        a.y = (tm && lab == kb + 1) ? 1.0f : 0.0f;
#pragma unroll
        for (int j = 0; j < 8; ++j) {
            v2f b;
            b.x = sh[kb * DIMF + j * 16 + n];
            b.y = sh[(kb + 1) * DIMF + j * 16 + n];
            acc[j] = __builtin_amdgcn_wmma_f32_16x16x4_f32(
                false, a, false, b, (short)0, acc[j], false, false);
        }
    }

    // D layout: VGPR r holds M=r (lanes 0-15) / M=r+8 (lanes 16-31), N = lane&15
#pragma unroll
    for (int j = 0; j < 8; ++j) {
#pragma unroll
        for (int r = 0; r < 8; ++r) {
            int row   = half ? (r + 8) : r;
            int onode = nodeBase + row;
            if (onode < N_NODES) {
                float v = acc[j][r];
                size_t off = (size_t)onode * DIMF + j * 16 + n;
                y0[off]  = v;
                out[off] = v;
            }
        }
    }
}

// ---------------- scatter: wave-per-edge, float4 per lane, f32 atomics into L2 ----------------
__global__ __launch_bounds__(256) void scatter_kernel(const int* __restrict__ src,
                                                      const int* __restrict__ dst,
                                                      const float* __restrict__ norm,
                                                      const float* __restrict__ out,
                                                      float* __restrict__ agg, int E) {
    int gt   = blockIdx.x * blockDim.x + threadIdx.x;
    int w    = gt >> 5;
    int lane = gt & 31;
    if (w >= E) return;

    // prefetch ahead on the HBM-streamed edge arrays (-> global_prefetch_b8)
    if (w + 512 < E) {
        __builtin_prefetch(&src[w + 512], 0, 0);
        __builtin_prefetch(&dst[w + 512], 0, 0);
        __builtin_prefetch(&norm[w + 512], 0, 0);
    }

    int   s  = src[w];
    int   d  = dst[w];
    float nm = norm[w];

    const float4* orow = (const float4*)(out + (size_t)s * DIMF); // 512B row, lane-coalesced
    float4 v = orow[lane];

    float* arow = agg + (size_t)d * DIMF + lane * 4;
    atomicAdd(arow + 0, v.x * nm);   // non-returning f32 atomics -> L2 atomic units
    atomicAdd(arow + 1, v.y * nm);
    atomicAdd(arow + 2, v.z * nm);
    atomicAdd(arow + 3, v.w * nm);
}

// ---------------- update: out = clamp(alpha*agg + (1-alpha)*y0, 0, 1) ----------------
__global__ void update_kernel(const float4* __restrict__ agg, const float4* __restrict__ y0,
                              const float* __restrict__ alpha_p, float4* __restrict__ out, int n4) {
    int i = blockIdx.x * blockDim.x + threadIdx.x;
    if (i >= n4) return;
    float al  = *alpha_p;
    float ral = 1.0f - al;
    float4 a = agg[i];
    float4 y = y0[i];
    float4 o;
    o.x = fminf(fmaxf(al * a.x + ral * y.x, 0.0f), 1.0f);
    o.y = fminf(fmaxf(al * a.y + ral * y.y, 0.0f), 1.0f);
    o.z = fminf(fmaxf(al * a.z + ral * y.z, 0.0f), 1.0f);
    o.w = fminf(fmaxf(al * a.w + ral * y.w, 0.0f), 1.0f);
    out[i] = o;
}

extern "C" void kernel_launch(void* const* d_in, const int* in_sizes, int n_in,
                              void* d_out, int out_size, void* d_ws, size_t ws_size,
                              hipStream_t stream) {
    const unsigned char* mask   = (const unsigned char*)d_in[0];  // bool [N] (1 byte)
    const float*         protos = (const float*)d_in[1];          // [C,D]
    const int*           labels = (const int*)d_in[2];            // [N]
    const int*           edge   = (const int*)d_in[3];            // [2,E]
    const float*         alpha  = (const float*)d_in[4];          // scalar
    float*               out    = (float*)d_out;                  // [N,D]

    const int* src = edge;             // edge_index[0]
    const int* dst = edge + N_EDGES;   // edge_index[1]

    // workspace layout (floats): deg[N] | norm[E] | y0[N*D] | agg[N*D]  (~109 MB)
    float* ws   = (float*)d_ws;
    float* deg  = ws;
    float* norm = deg + N_NODES;
    float* y0   = norm + N_EDGES;
    float* agg  = y0 + (size_t)N_NODES * DIMF;

    const int TB = 256;   // 8 wave32 waves per block

    zero_f4<<<(N_NODES / 4 + TB - 1) / TB, TB, 0, stream>>>((float4*)deg, N_NODES / 4);
    deg_kernel<<<(N_EDGES + TB - 1) / TB, TB, 0, stream>>>(dst, deg, N_EDGES);
    dis_kernel<<<(N_NODES + TB - 1) / TB, TB, 0, stream>>>(deg, N_NODES);
    norm_kernel<<<(N_EDGES + TB - 1) / TB, TB, 0, stream>>>(src, dst, deg, norm, N_EDGES);

    seed_wmma<<<(N_NODES + 127) / 128, 256, 0, stream>>>(mask, protos, labels, y0, out);

    const int nd4 = (N_NODES * DIMF) / 4;  // 3.2M float4
    for (int l = 0; l < LAYERS; ++l) {
        zero_f4<<<(nd4 + TB - 1) / TB, TB, 0, stream>>>((float4*)agg, nd4);
        scatter_kernel<<<(N_EDGES * 32 + TB - 1) / TB, TB, 0, stream>>>(src, dst, norm, out, agg, N_EDGES);
        update_kernel<<<(nd4 + TB - 1) / TB, TB, 0, stream>>>((const float4*)agg, (const float4*)y0,
                                                              alpha, (float4*)out, nd4);
    }
}